// Motion_prediction_7962869367083
// MI455X (gfx1250) — compile-verified
//
#include <hip/hip_runtime.h>
#include <hip/hip_bf16.h>
#include <stdint.h>

// ---------------------------------------------------------------------------
// Motion prediction (ragged CAV batch) for MI455X / gfx1250.
//
// Memory-bound problem: ~136 MB of HBM traffic, ~0.18 flop/byte -> WMMA is
// roofline-irrelevant; we optimize the data-movement path instead:
//   * one wave32 per CAV (counts <= 31) -> no per-object searchsorted
//   * gfx1250 async global->LDS staging of per-block offsets/time rows
//     (GLOBAL_LOAD_ASYNC_TO_LDS_B32 + s_wait_asynccnt), with sync fallback
//   * global_prefetch_b8 hint for the block's bbx region
//   * vectorized b128/b64 stores for the outputs
// ---------------------------------------------------------------------------

#define CPB 8  // CAVs per 256-thread block (one wave32 each)

#if defined(__has_builtin)
#if __has_builtin(__builtin_amdgcn_global_load_async_to_lds_b32)
#define HAVE_ASYNC_LDS 1
#endif
#endif

__device__ __forceinline__ void async_copy_b32(const void* g, void* l) {
#if defined(HAVE_ASYNC_LDS)
  __builtin_amdgcn_global_load_async_to_lds_b32(
      (__attribute__((address_space(1))) int*)(void*)g,
      (__attribute__((address_space(3))) int*)l,
      /*offset=*/0, /*cpol=*/0);
#else
  *(volatile int*)l = *(const int*)g;
#endif
}

__device__ __forceinline__ void wait_async_all() {
#if defined(HAVE_ASYNC_LDS)
#if __has_builtin(__builtin_amdgcn_s_wait_asynccnt)
  __builtin_amdgcn_s_wait_asynccnt(0);
#else
  asm volatile("s_wait_asynccnt 0" ::: "memory");
#endif
#endif
}

// ---------------------------------------------------------------------------
// Kernel 1: inclusive prefix sum of per-CAV counts (single block).
// Each thread serially sums a contiguous chunk, one LDS Hillis-Steele scan of
// the 1024 partials, then each thread emits its chunk's inclusive results.
// ---------------------------------------------------------------------------
__global__ __launch_bounds__(1024) void scan_kernel(const int* __restrict__ counts,
                                                    int* __restrict__ offsets,
                                                    int n) {
  __shared__ int sh[1024];
  const int tid = threadIdx.x;
  const int chunk = (n + 1023) >> 10;
  int begin = tid * chunk;
  int end = begin + chunk;
  if (begin > n) begin = n;
  if (end > n) end = n;

  int sum = 0;
  for (int i = begin; i < end; ++i) sum += counts[i];
  sh[tid] = sum;
  __syncthreads();

  for (int off = 1; off < 1024; off <<= 1) {
    int add = (tid >= off) ? sh[tid - off] : 0;
    __syncthreads();
    sh[tid] += add;
    __syncthreads();
  }

  int run = (tid == 0) ? 0 : sh[tid - 1];  // exclusive prefix of this chunk
  for (int i = begin; i < end; ++i) {
    run += counts[i];
    offsets[i] = run;  // inclusive cumsum
  }
}

// ---------------------------------------------------------------------------
// Kernel 2: one wave32 per CAV. Faithful FP-order reproduction of the
// reference (coords normalized before differencing; exact where-guards).
//
// Output layout in d_out (float32), N = n_obj:
//   [0        , 8N )  obj_input        [1,N,2,4]
//   [8N       , 10N)  query            [1,N,1,2]
//   [10N      , 12N)  time_encode      [N,2]
//   [12N      , 13N)  target_time_diff [N]
// ---------------------------------------------------------------------------
__global__ __launch_bounds__(32 * CPB) void motion_kernel(
    const float* __restrict__ bbx,      // [N,3,7]
    const float* __restrict__ tim,      // [C*3]
    const int* __restrict__ offsets,    // [C] inclusive cumsum of counts
    float* __restrict__ out,
    int num_cav, int n_obj) {
  __shared__ int sh_prev[CPB];
  __shared__ int sh_end[CPB];
  __shared__ float sh_time[CPB * 3];

  const int lane = threadIdx.x & 31;
  const int warp = threadIdx.x >> 5;
  const int cbase = blockIdx.x * CPB;

  // Wave 0 stages the block's offset window and time rows into LDS via the
  // gfx1250 async-to-LDS path, then drains ASYNCcnt before the barrier.
  if (warp == 0) {
    int gi = cbase - 1 + lane;  // offsets[c-1] for each CAV in block
    if (lane < CPB && gi >= 0 && gi < num_cav)
      async_copy_b32(offsets + gi, &sh_prev[lane]);
    int gj = cbase + lane;      // offsets[c]
    if (lane < CPB && gj < num_cav)
      async_copy_b32(offsets + gj, &sh_end[lane]);
    int gk = cbase * 3 + lane;  // time rows (3 floats per CAV)
    if (lane < 3 * CPB && gk < 3 * num_cav)
      async_copy_b32(tim + gk, &sh_time[lane]);
    wait_async_all();
  }
  __syncthreads();

  const int c = cbase + warp;
  if (c >= num_cav) return;

  const int start = (c == 0) ? 0 : sh_prev[warp];
  const int end = sh_end[warp];

  // td_rep = [t2, t1, t0] (flipped time row of this CAV)
  const float t0 = sh_time[warp * 3 + 0];
  const float t1 = sh_time[warp * 3 + 1];
  const float t2 = sh_time[warp * 3 + 2];

  const float dt0 = t1 - t2;            // time_encode[:,0]
  const float dt1 = t0 - t1;            // time_encode[:,1]
  const float tn1 = t1 - t0;            // td_norm[:,-2]
  const float last_len = 0.0f - tn1;    // td_norm[:,-1] - td_norm[:,-2]
  const bool len_zero = (last_len == 0.0f);
  const float ttd = -t0;                // target_time_diff

  const size_t N = (size_t)n_obj;
  float* __restrict__ q_out = out + 8 * N;
  float* __restrict__ te_out = out + 10 * N;
  float* __restrict__ tt_out = out + 12 * N;

  if (lane == 0) __builtin_prefetch(bbx + (size_t)start * 21);  // global_prefetch_b8

  for (int n = start + lane; n < end; n += 32) {
    const float* __restrict__ b = bbx + (size_t)n * 21;
    const float b0x = b[0], b0y = b[1];     // frame 0 (becomes coords[-1])
    const float b1x = b[7], b1y = b[8];     // frame 1
    const float b2x = b[14], b2y = b[15];   // frame 2 (becomes coords[0])

    // coords (flipped, normalized to last): p0 = b2-b0, p1 = b1-b0, p2 = b0-b0
    const float p0x = b2x - b0x, p0y = b2y - b0y;
    const float p1x = b1x - b0x, p1y = b1y - b0y;
    const float p2x = b0x - b0x, p2y = b0y - b0y;  // exact +0 for finite input

    const float d0x = p1x - p0x, d0y = p1y - p0y;  // distance row 0
    const float d1x = p2x - p1x, d1y = p2y - p1y;  // distance row 1
    const float s0x = d0x / dt0, s0y = d0y / dt0;  // speed row 0
    const float s1x = d1x / dt1, s1y = d1y / dt1;  // speed row 1

    float4* __restrict__ oi = (float4*)(out + (size_t)n * 8);  // 32B aligned
    oi[0] = make_float4(d0x, d0y, s0x, s0y);
    oi[1] = make_float4(d1x, d1y, s1x, s1y);

    // extrap = p2 + (p2 - p1) * (-t0) / last_len ; query = where(len==0, 0, extrap)
    float qx, qy;
    if (len_zero) {
      qx = 0.0f;
      qy = 0.0f;
    } else {
      qx = p2x + ((p2x - p1x) * (-t0)) / last_len;
      qy = p2y + ((p2y - p1y) * (-t0)) / last_len;
    }
    ((float2*)q_out)[n] = make_float2(qx, qy);
    ((float2*)te_out)[n] = make_float2(dt0, dt1);
    tt_out[n] = ttd;
  }
}

// ---------------------------------------------------------------------------
// Launcher. Inputs (setup_inputs order):
//   d_in[0]: past_k_object_bbx   float32 [N,3,7]   (in_sizes[0] = 21*N)
//   d_in[1]: past_k_time_interval float32 [3*C]
//   d_in[2]: past_k_object_cav_num int32  [C]
// d_ws: offsets int32[C] (C*4 = 248 KB)
// ---------------------------------------------------------------------------
extern "C" void kernel_launch(void* const* d_in, const int* in_sizes, int n_in,
                              void* d_out, int out_size, void* d_ws, size_t ws_size,
                              hipStream_t stream) {
  (void)n_in;
  (void)out_size;
  (void)ws_size;
  const float* bbx = (const float*)d_in[0];
  const float* tim = (const float*)d_in[1];
  const int* counts = (const int*)d_in[2];
  float* out = (float*)d_out;

  const int num_cav = in_sizes[2];
  const int n_obj = in_sizes[0] / 21;  // [N,3,7]
  int* offsets = (int*)d_ws;

  hipLaunchKernelGGL(scan_kernel, dim3(1), dim3(1024), 0, stream,
                     counts, offsets, num_cav);

  const int blocks = (num_cav + CPB - 1) / CPB;
  hipLaunchKernelGGL(motion_kernel, dim3(blocks), dim3(32 * CPB), 0, stream,
                     bbx, tim, offsets, out, num_cav, n_obj);
}